// Proto_13589276525295
// MI455X (gfx1250) — compile-verified
//
#include <hip/hip_runtime.h>
#include <math.h>

typedef float v2f __attribute__((ext_vector_type(2)));
typedef float v8f __attribute__((ext_vector_type(8)));
typedef unsigned int v4u __attribute__((ext_vector_type(4)));
typedef int v4i __attribute__((ext_vector_type(4)));
typedef int v8i __attribute__((ext_vector_type(8)));

#define HID   768
#define PSZ   128
#define BATCH 32
#define NREL  16
#define KR    (BATCH*NREL)      // 512
#define NENT  (KR*PSZ)          // 65536
#define HTILE 128
#define GC    64
#define NCH   (HID/GC)          // 12
#define MPAD  4                 // +4 dwords pad per 64-dword row (16B aligned)

// ---- workspace layout (float offsets) ----
#define OFF_CNT    0            // [512*128] f32 histogram counts
#define OFF_PROP   65536        // [32*768]
#define OFF_NK     90112        // [32]
#define OFF_BSUM   90144        // [32]
#define OFF_TOT    90176        // [1]
#define OFF_LACC   90177        // [1]
#define ZERO_N     90178        // zero-init region length
#define OFF_HM     90180        // [65536] int (head or -1)
#define OFF_W      155716       // [512*768]
#define OFF_V      548932       // [512*768]
#define OFF_S      942148       // [512*128] sigmoid'd sims over all protos
#define OFF_ROWSUM 1007684      // [512]
#define OFF_LU     1008196      // [32] int labels_unique
#define OFF_INV    1008228      // [32] int inverse map
#define OFF_LUPOS  1008260      // [128] int proto-row -> k (or -1)

__device__ __forceinline__ v8f wmma_f32(v2f a, v2f b, v8f c) {
    return __builtin_amdgcn_wmma_f32_16x16x4_f32(false, a, false, b, (short)0, c,
                                                 false, false);
}

// Generic->LDS offset: low 32 bits of the aperture-based generic address are
// the wave-relative LDS byte offset (CDNA5 flat aperture rule).
__device__ __forceinline__ unsigned lds_off(const void* p) {
    return (unsigned)(unsigned long long)(uintptr_t)p;
}

// Issue a TDM load of a 128x64 f32 tile (row stride 768 elems in memory) into
// LDS at `ldsa`, with +4-dword padding after every 64 dwords (LDS row stride
// 68 dwords).  2D descriptor; groups 2/3 filled with benign dims.
__device__ __forceinline__ void tdm_load_tile(const float* gptr, unsigned ldsa) {
    unsigned long long ga = (unsigned long long)(uintptr_t)gptr;
    v4u g0 = { 1u,                                  // count=1 valid descriptor
               ldsa,                                // lds_addr
               (unsigned)ga,                        // global_addr[31:0]
               (unsigned)((ga >> 32) & 0x1FFFFFFull) | 0x80000000u }; // type=2
    // group1: data_size=4B(2), pad_enable, pad_interval=64dw(5), pad_amount=4dw(3)
    int w0 = (2 << 16) | (1 << 20) | (5 << 22) | (3 << 25);
    v8i g1 = { w0,
               (int)((HID & 0xFFFF) << 16),         // tensor_dim0 lo16
               (int)((HID >> 16) | ((HID & 0xFFFF) << 16)), // dim0 hi | dim1 lo
               (int)((HID >> 16) | (GC << 16)),     // dim1 hi | tile_dim0=64
               (int)HTILE,                          // tile_dim1=128, tile_dim2=0
               (int)HID,                            // tensor_dim0_stride lo32
               (int)(((HID * HID) & 0xFFFF) << 16), // dim1_stride lo16
               (int)((HID * HID) >> 16) };          // dim1_stride hi
    v4i g2 = { 1, 1, 0, 0 };                        // tensor_dim2=1, tensor_dim3=1
    v4i g3 = { 0, 1 << 16, 0, 0 };                  // tensor_dim4=1
    v8i g4 = { 0, 0, 0, 0, 0, 0, 0, 0 };            // unused trailing group
    __builtin_amdgcn_tensor_load_to_lds(g0, g1, g2, g3, g4, 0);
}

// -------- init: zero the accumulated regions --------
__global__ void k_init(float* ws) {
    int i = blockIdx.x * 256 + threadIdx.x;
    if (i < ZERO_N) ws[i] = 0.0f;
}

// -------- A1: sort labels (rank), inverse map, proto-row position map --------
__global__ void k_labels(const int* __restrict__ labels,
                         int* __restrict__ lu, int* __restrict__ inv,
                         int* __restrict__ lupos) {
    __shared__ int lab[BATCH];
    int t = threadIdx.x;                       // 128 threads
    if (t < BATCH) lab[t] = labels[t];
    if (t < PSZ) lupos[t] = -1;
    __syncthreads();
    if (t < BATCH) {
        int mine = lab[t], rank = 0;
        for (int j = 0; j < BATCH; j++) {
            int lj = lab[j];
            rank += (lj < mine) || (lj == mine && j < t);
        }
        inv[t] = rank;
        lu[rank] = mine;
        lupos[mine] = rank;                    // labels distinct, rows distinct
    }
}

// -------- A2a: row sums / batch sums / total over rel_k --------
__global__ __launch_bounds__(128) void k_sums(const int* __restrict__ rel,
                                              const int* __restrict__ inv,
                                              float* __restrict__ rowsum,
                                              float* __restrict__ bsum,
                                              float* __restrict__ tot) {
    __shared__ float red[128];
    int kr = blockIdx.x, k = kr >> 4, r = kr & 15, p = threadIdx.x;
    int id = rel[(inv[k] * NREL + r) * PSZ + p];
    red[p] = (float)id;
    __syncthreads();
    for (int s = 64; s > 0; s >>= 1) {
        if (p < s) red[p] += red[p + s];
        __syncthreads();
    }
    if (p == 0) {
        rowsum[kr] = red[0];
        atomicAdd(&bsum[k], red[0]);
        atomicAdd(&tot[0], red[0]);
    }
}

// -------- A2b: mask, head indices, histogram counts, n[k] --------
__global__ __launch_bounds__(128) void k_mask(const int* __restrict__ rel,
                                              const int* __restrict__ inv,
                                              const float* __restrict__ rowsum,
                                              const float* __restrict__ bsum,
                                              const float* __restrict__ tot,
                                              int* __restrict__ hm,
                                              float* __restrict__ cnt,
                                              float* __restrict__ nk) {
    __shared__ int bins[128];
    int kr = blockIdx.x, k = kr >> 4, r = kr & 15, p = threadIdx.x;
    bins[p] = 0;
    __syncthreads();
    int id = rel[(inv[k] * NREL + r) * PSZ + p];
    bool m = (id != PSZ) && (rowsum[kr] > 0.0f) && (bsum[k] > 0.0f) &&
             (tot[0] > 0.0f);
    int head = id < (PSZ - 1) ? id : (PSZ - 1);
    hm[kr * PSZ + p] = m ? head : -1;
    if (m) atomicAdd(&bins[head], 1);
    __syncthreads();
    cnt[kr * PSZ + p] = (float)bins[p];
    __syncthreads();
    for (int s = 64; s > 0; s >>= 1) {
        if (p < s) bins[p] += bins[p + s];
        __syncthreads();
    }
    if (p == 0) atomicAdd(&nk[k], (float)bins[0]);
}

// -------- B: w[kr,h] = cnt[kr,:] @ P0   (WMMA f32 16x16x4, K=128) --------
__global__ __launch_bounds__(32) void k_w(const float* __restrict__ cnt,
                                          const float* __restrict__ protos,
                                          float* __restrict__ wbuf) {
    int krt = blockIdx.x & 31;        // 32 kr-tiles
    int ht  = blockIdx.x >> 5;        // 48 h-tiles
    int lane = threadIdx.x;
    int l15 = lane & 15;
    bool hi = lane >= 16;
    int mrow = krt * 16 + l15;
    int ncol = ht * 16 + l15;
    v8f acc = {};
    for (int pp = 0; pp < PSZ; pp += 4) {
        int pa = pp + (hi ? 2 : 0);
        v2f a = { cnt[mrow * PSZ + pa], cnt[mrow * PSZ + pa + 1] };
        v2f b = { protos[pa * HID + ncol], protos[(pa + 1) * HID + ncol] };
        acc = wmma_f32(a, b, acc);
    }
    int mb = hi ? 8 : 0;
#pragma unroll
    for (int i = 0; i < 8; i++)
        wbuf[(krt * 16 + mb + i) * HID + ncol] = acc[i];
}

// -------- C: single TDM-fed streaming pass over relation_embedding --------
// Per block: one (r, h-block of 128). Double-buffered tensor_load_to_lds of
// 128x64 M tiles; computes v[k,r,h] = pt @ M_r^T for its h-slice and
// atomically accumulates prop[k,g] += w[k,r,h-slice] @ M_r.
__global__ __launch_bounds__(256) void k_fused(const float* __restrict__ relM,
                                               const float* __restrict__ protos,
                                               const int* __restrict__ lu,
                                               const float* __restrict__ wbuf,
                                               float* __restrict__ vbuf,
                                               float* __restrict__ prop) {
    __shared__ float Ms[2][HTILE][GC + MPAD];   // 2 x 34816 B (TDM pad = 4 dw)
    __shared__ float wk[BATCH][HTILE + MPAD];   // 16896 B
    int r  = blockIdx.x & 15;
    int hb = blockIdx.x >> 4;                   // 0..5
    int h0 = hb * HTILE;
    int t = threadIdx.x, wave = t >> 5, lane = t & 31;
    int l15 = lane & 15;
    bool hi = lane >= 16;

    for (int i = t; i < BATCH * HTILE; i += 256)
        wk[i / HTILE][i % HTILE] =
            wbuf[((i / HTILE) * NREL + r) * HID + h0 + (i % HTILE)];

    const float* tile_base = relM + (size_t)r * HID * HID + (size_t)h0 * HID;
    const float* pt0 = protos + (size_t)lu[l15] * HID;        // shared A rows
    const float* pt1 = protos + (size_t)lu[16 + l15] * HID;

    v8f accV0 = {}, accV1 = {};
    int kt = wave >> 2, gt = wave & 3;          // prop tile assignment

    if (wave == 0) tdm_load_tile(tile_base, lds_off(&Ms[0][0][0]));

    for (int c = 0; c < NCH; ++c) {
        int cur = c & 1;
        if (wave == 0) {
            if (c + 1 < NCH) {
                tdm_load_tile(tile_base + (c + 1) * GC,
                              lds_off(&Ms[cur ^ 1][0][0]));
                __builtin_amdgcn_s_wait_tensorcnt(1);   // tile c complete
            } else {
                __builtin_amdgcn_s_wait_tensorcnt(0);
            }
        }
        __syncthreads();                         // publish Ms[cur] to block

        int g0 = c * GC;
        // ---- V: D[16k x 16h] += pt[16k x 4g] * M^T[4g x 16h]; wave owns htile
        for (int gg = 0; gg < GC; gg += 4) {
            int ca = gg + (hi ? 2 : 0);
            int ga = g0 + ca;
            v2f a0 = { pt0[ga], pt0[ga + 1] };
            v2f a1 = { pt1[ga], pt1[ga + 1] };
            v2f b  = { Ms[cur][wave * 16 + l15][ca],
                       Ms[cur][wave * 16 + l15][ca + 1] };
            accV0 = wmma_f32(a0, b, accV0);
            accV1 = wmma_f32(a1, b, accV1);
        }

        // ---- prop: D[16k x 16g] += w[16k x 4h] * M[4h x 16g]
        v8f accP = {};
        for (int hh = 0; hh < HTILE; hh += 4) {
            int ha = hh + (hi ? 2 : 0);
            v2f a = { wk[kt * 16 + l15][ha], wk[kt * 16 + l15][ha + 1] };
            v2f b = { Ms[cur][ha][gt * 16 + l15], Ms[cur][ha + 1][gt * 16 + l15] };
            accP = wmma_f32(a, b, accP);
        }
        int ncol = g0 + gt * 16 + l15;
        int mb = kt * 16 + (hi ? 8 : 0);
#pragma unroll
        for (int i = 0; i < 8; i++)
            atomicAdd(&prop[(mb + i) * HID + ncol], accP[i]);

        __syncthreads();                         // done reading Ms[cur]
    }

    int hcol = h0 + wave * 16 + l15;
    int mb = hi ? 8 : 0;
#pragma unroll
    for (int i = 0; i < 8; i++) {
        vbuf[((mb + i) * NREL + r) * HID + hcol]      = accV0[i];
        vbuf[((16 + mb + i) * NREL + r) * HID + hcol] = accV1[i];
    }
}

// -------- D: S[kr,p'] = sigmoid(v[kr,:] . P0[p',:])  (WMMA, K=768) --------
__global__ __launch_bounds__(32) void k_sim(const float* __restrict__ vbuf,
                                            const float* __restrict__ protos,
                                            float* __restrict__ S) {
    int krt = blockIdx.x & 31;        // 32 kr-tiles
    int pt_ = blockIdx.x >> 5;        // 8 p-tiles
    int lane = threadIdx.x, l15 = lane & 15;
    bool hi = lane >= 16;
    int mrow = krt * 16 + l15;
    int ncol = pt_ * 16 + l15;
    v8f acc = {};
    for (int hh = 0; hh < HID; hh += 4) {
        int ha = hh + (hi ? 2 : 0);
        v2f a = { vbuf[mrow * HID + ha], vbuf[mrow * HID + ha + 1] };
        v2f b = { protos[ncol * HID + ha], protos[ncol * HID + ha + 1] };
        acc = wmma_f32(a, b, acc);
    }
    int mb = hi ? 8 : 0;
#pragma unroll
    for (int i = 0; i < 8; i++) {
        float x = acc[i];
        S[(krt * 16 + mb + i) * PSZ + ncol] = 1.0f / (1.0f + expf(-x));
    }
}

// -------- E: cross-entropy accumulation over 65536 entries --------
__global__ __launch_bounds__(256) void k_loss(const int* __restrict__ hm,
                                              const float* __restrict__ S,
                                              float* __restrict__ lacc) {
    __shared__ float red[256];
    int e = blockIdx.x * 256 + threadIdx.x;
    int h = hm[e];
    float term;
    if (h >= 0) {
        float s = S[(e >> 7) * PSZ + h];      // active: l=(1-s,s), label 1
        term = log1pf(expf(1.0f - 2.0f * s));
    } else {
        term = 0.69314718055994531f;          // both (0,0) and (1,1) -> ln 2
    }
    red[threadIdx.x] = term;
    __syncthreads();
    for (int s = 128; s > 0; s >>= 1) {
        if (threadIdx.x < s) red[threadIdx.x] += red[threadIdx.x + s];
        __syncthreads();
    }
    if (threadIdx.x == 0) atomicAdd(lacc, red[0]);
}

// -------- F: proto_embedding scatter-update + final loss value --------
__global__ __launch_bounds__(256) void k_proto(const float* __restrict__ protos,
                                               const float* __restrict__ prop,
                                               const float* __restrict__ nk,
                                               const int* __restrict__ lupos,
                                               const float* __restrict__ tot,
                                               const float* __restrict__ lacc,
                                               float* __restrict__ out) {
    int p = blockIdx.x;
    int kk = lupos[p];
    float nv = (kk >= 0) ? nk[kk] : 0.0f;
    bool upd = (kk >= 0) && (nv > 0.0f);
    float inv_n = 1.0f / fmaxf(nv, 1.0f);
    float* pe = out + BATCH * PSZ + 1;        // proto_embedding region
    for (int h = threadIdx.x; h < HID; h += 256) {
        float old = protos[p * HID + h];
        pe[p * HID + h] =
            upd ? 0.5f * old + 0.5f * prop[kk * HID + h] * inv_n : old;
    }
    if (p == 0 && threadIdx.x == 0)
        out[BATCH * PSZ] = (tot[0] > 0.0f) ? lacc[0] / (float)NENT : 0.0f;
}

// -------- G: logits = -||instance - proto_embedding||^2 --------
__global__ __launch_bounds__(128) void k_logits(const float* __restrict__ inst,
                                                float* __restrict__ out) {
    __shared__ float xb[HID];
    int b = blockIdx.x;
    for (int h = threadIdx.x; h < HID; h += 128) xb[h] = inst[b * HID + h];
    __syncthreads();
    const float* pe = out + BATCH * PSZ + 1 + (size_t)threadIdx.x * HID;
    float acc = 0.0f;
    for (int h = 0; h < HID; h++) {
        float d = xb[h] - pe[h];
        acc = fmaf(d, d, acc);
    }
    out[b * PSZ + threadIdx.x] = -acc;
}

extern "C" void kernel_launch(void* const* d_in, const int* in_sizes, int n_in,
                              void* d_out, int out_size, void* d_ws, size_t ws_size,
                              hipStream_t stream) {
    const float* inst  = (const float*)d_in[0];
    const float* relM  = (const float*)d_in[1];
    const int*   rel   = (const int*)d_in[2];
    const int*   labs  = (const int*)d_in[3];
    const float* proto = (const float*)d_in[4];
    float* out = (float*)d_out;
    float* ws  = (float*)d_ws;

    float* cnt    = ws + OFF_CNT;
    float* prop   = ws + OFF_PROP;
    float* nk     = ws + OFF_NK;
    float* bsum   = ws + OFF_BSUM;
    float* tot    = ws + OFF_TOT;
    float* lacc   = ws + OFF_LACC;
    int*   hm     = (int*)(ws + OFF_HM);
    float* wbuf   = ws + OFF_W;
    float* vbuf   = ws + OFF_V;
    float* S      = ws + OFF_S;
    float* rowsum = ws + OFF_ROWSUM;
    int*   lu     = (int*)(ws + OFF_LU);
    int*   inv    = (int*)(ws + OFF_INV);
    int*   lupos  = (int*)(ws + OFF_LUPOS);

    k_init  <<<(ZERO_N + 255) / 256, 256, 0, stream>>>(ws);
    k_labels<<<1, 128, 0, stream>>>(labs, lu, inv, lupos);
    k_sums  <<<KR, 128, 0, stream>>>(rel, inv, rowsum, bsum, tot);
    k_mask  <<<KR, 128, 0, stream>>>(rel, inv, rowsum, bsum, tot, hm, cnt, nk);
    k_w     <<<32 * 48, 32, 0, stream>>>(cnt, proto, wbuf);
    k_fused <<<NREL * (HID / HTILE), 256, 0, stream>>>(relM, proto, lu, wbuf,
                                                       vbuf, prop);
    k_sim   <<<32 * 8, 32, 0, stream>>>(vbuf, proto, S);
    k_loss  <<<NENT / 256, 256, 0, stream>>>(hm, S, lacc);
    k_proto <<<PSZ, 256, 0, stream>>>(proto, prop, nk, lupos, tot, lacc, out);
    k_logits<<<BATCH, 128, 0, stream>>>(inst, out);
}